// DistMaps_87316685128420
// MI455X (gfx1250) — compile-verified
//
#include <hip/hip_runtime.h>
#include <hip/hip_bf16.h>

typedef __attribute__((ext_vector_type(2))) float v2f;
typedef __attribute__((ext_vector_type(8))) float v8f;

#define BATCH      4
#define HDIM       512
#define WDIM       512
#define HW         (HDIM * WDIM)
#define NPTS       24            // points per polarity group
#define EMB        12
#define RAD2       25.0f         // (NORM_RADIUS*SPATIAL_SCALE)^2
#define INVALID_D  1.0e6f
#define PAD_D      2.0e8f

#define WAVES_PER_BLOCK 8
#define TILES_PER_WAVE  8
// tiles/batch = HW/16 = 16384 ; waves/batch = 16384/8 = 2048 ; blocks/batch = 256

__global__ __launch_bounds__(256) void distmaps_wmma_kernel(
    const float* __restrict__ coords,      // [B, 48, 3] (row, col, order)
    const float* __restrict__ embed_table, // [50, 12]
    float* __restrict__ out)               // [B, 26, H, W]
{
    __shared__ __align__(16) float s_embed[50 * EMB]; // 2400 B, L2->LDS once per block
    const int tid = threadIdx.x;
    for (int i = tid; i < 50 * EMB; i += 256) s_embed[i] = embed_table[i];
    __syncthreads();

    const int lane        = tid & 31;
    const int waveInBlock = tid >> 5;
    const int gw          = blockIdx.x * WAVES_PER_BLOCK + waveInBlock;
    const int wavesPerBatch = (HW / 16) / TILES_PER_WAVE; // 2048
    const int b     = gw / wavesPerBatch;
    const int wslot = gw % wavesPerBatch;
    const int pix0  = wslot * (16 * TILES_PER_WAVE);      // 128 consecutive pixels, row-aligned

    const int  n  = lane & 15;
    const bool hi = lane >= 16;

    // ---- A matrices: points, 16x4 f32. Lanes 0-15 hold (K0,K1)=(-2r,-2c) for
    // point M=lane; lanes 16-31 hold (K2,K3)=(r^2+c^2, 1) for point M=lane-16.
    // Invalid point -> column [0,0,1e6,0]; pad (j>=24) -> [0,0,2e8,0].
    v2f A[2][2];
#pragma unroll
    for (int g = 0; g < 2; ++g) {
#pragma unroll
        for (int t = 0; t < 2; ++t) {
            const int j = t * 16 + n;
            float ax, ay;
            if (j < NPTS) {
                const float* p = coords + ((size_t)b * 48 + (size_t)g * NPTS + j) * 3;
                const float r = p[0], c = p[1];
                const bool inval = fmaxf(r, c) < 0.0f;
                if (hi) { ax = inval ? INVALID_D : (r * r + c * c); ay = inval ? 0.0f : 1.0f; }
                else    { ax = inval ? 0.0f : (-2.0f * r);          ay = inval ? 0.0f : (-2.0f * c); }
            } else {
                ax = hi ? PAD_D : 0.0f;
                ay = 0.0f;
            }
            A[g][t].x = ax;
            A[g][t].y = ay;
        }
    }

    const size_t outB   = (size_t)b * 26 * HW;
    const int    g_lane = hi ? 1 : 0;

    for (int k = 0; k < TILES_PER_WAVE; ++k) {
        const int   pix   = pix0 + k * 16;   // tile base pixel (never crosses an h-row)
        const int   h     = pix >> 9;        // / 512
        const int   wbase = pix & 511;
        const float hN    = (float)h;
        const float wN    = (float)(wbase + n);

        // B matrix: pixels, 4x16 f32. VGPR0 = rows {K0,K2}, VGPR1 = rows {K1,K3}
        // (mirrors the documented A 16x4 half-split). Column N = lane&15.
        v2f Bm;
        Bm.x = hi ? 1.0f : hN;                 // K2 = 1        | K0 = h
        Bm.y = hi ? (hN * hN + wN * wN) : wN;  // K3 = h^2+w^2  | K1 = w

        float dmin[2];
        int   didx[2];
#pragma unroll
        for (int g = 0; g < 2; ++g) {
            v8f c0 = {0.f, 0.f, 0.f, 0.f, 0.f, 0.f, 0.f, 0.f};
            // D[point M, pixel N] = d^2 ; two subtiles cover local point idx 0-15, 16-31
            v8f d0 = __builtin_amdgcn_wmma_f32_16x16x4_f32(
                false, A[g][0], false, Bm, (short)0, c0, false, false);
            v8f d1 = __builtin_amdgcn_wmma_f32_16x16x4_f32(
                false, A[g][1], false, Bm, (short)0, c0, false, false);

            // In-lane first-argmin over increasing local index (strict <):
            // lanes 0-15 see rows {0..7,16..23}, lanes 16-31 see {8..15,24..31}.
            const int base = hi ? 8 : 0;
            float bv = d0[0];
            int   bi = base;
#pragma unroll
            for (int v = 1; v < 8; ++v) {
                const float x = d0[v];
                const bool  t = x < bv;
                bv = t ? x : bv;
                bi = t ? (base + v) : bi;
            }
#pragma unroll
            for (int v = 0; v < 8; ++v) {
                const float x = d1[v];
                const bool  t = x < bv;
                bv = t ? x : bv;
                bi = t ? (base + 16 + v) : bi;
            }
            // Cross-half combine (indices interleave -> explicit tie-break on idx)
            const float ov = __shfl_xor(bv, 16, 32);
            const int   oi = __shfl_xor(bi, 16, 32);
            const bool  tk = (ov < bv) || (ov == bv && oi < bi);
            dmin[g] = tk ? ov : bv;
            didx[g] = tk ? oi : bi;
        }

        // Lanes 0-15 emit group 0, lanes 16-31 emit group 1 for pixel pix+n.
        const float dm = dmin[g_lane];
        const int   id = didx[g_lane];
        const bool  on = dm <= RAD2;
        const float disk = on ? 1.0f : 0.0f;
        const int   pos  = on ? (id + 1) : 0;
        const int   lpix = pix + n;

        out[outB + (size_t)g_lane * HW + lpix] = disk;

        // emb: 12 contiguous floats at 2*HW + (g*HW + pix)*12 (48B, 16B-aligned)
        const float4* erow = (const float4*)(s_embed + pos * EMB);
        float4* dst = (float4*)(out + outB + 2 * (size_t)HW +
                                ((size_t)g_lane * HW + (size_t)lpix) * EMB);
        dst[0] = erow[0];
        dst[1] = erow[1];
        dst[2] = erow[2];
    }
}

extern "C" void kernel_launch(void* const* d_in, const int* in_sizes, int n_in,
                              void* d_out, int out_size, void* d_ws, size_t ws_size,
                              hipStream_t stream) {
    (void)in_sizes; (void)n_in; (void)d_ws; (void)ws_size; (void)out_size;
    // d_in[0] = x [4,3,512,512] (shape-only in the reference; values unused)
    const float* coords      = (const float*)d_in[1]; // [4,48,3]
    const float* embed_table = (const float*)d_in[2]; // [50,12]
    float*       out         = (float*)d_out;         // [4,26,512,512]

    const int blocksPerBatch = ((HW / 16) / TILES_PER_WAVE) / WAVES_PER_BLOCK; // 256
    dim3 grid(BATCH * blocksPerBatch); // 1024 blocks * 256 thr = exact cover, EXEC all-1s
    dim3 block(256);
    distmaps_wmma_kernel<<<grid, block, 0, stream>>>(coords, embed_table, out);
}